// Attention_72877005078652
// MI455X (gfx1250) — compile-verified
//
#include <hip/hip_runtime.h>
#include <cfloat>
#include <cmath>

#define B_   64
#define T_   2048
#define RNN_ 1024
#define EMB_ 512
#define ATT_ 128
#define NF_  32
#define KS_  31
#define PAD_ 15

typedef __attribute__((ext_vector_type(2))) float v2f;
typedef __attribute__((ext_vector_type(8))) float v8f;
typedef __attribute__((ext_vector_type(4))) unsigned int u32x4;
typedef __attribute__((ext_vector_type(8))) int i32x8;
typedef __attribute__((ext_vector_type(4))) int i32x4;

// ---------------------------------------------------------------------------
// Kernel 1: pq = attention_hidden_state (64x1024) @ Wq^T (1024x128) -> (64,128)
// One wave per 16x16 output tile; V_WMMA_F32_16X16X4_F32 over K=1024.
// ---------------------------------------------------------------------------
__global__ __launch_bounds__(32) void k_pq(const float* __restrict__ hid,
                                           const float* __restrict__ Wq,
                                           float* __restrict__ pq) {
    const int mtile = blockIdx.x & 3;   // 64/16 = 4 M tiles
    const int ntile = blockIdx.x >> 2;  // 128/16 = 8 N tiles
    const int lane  = threadIdx.x;
    const int half  = lane >> 4;
    const int l     = lane & 15;
    const int M     = mtile * 16 + l;
    const int N     = ntile * 16 + l;
    const int koff  = half * 2;

    v8f c = {};
    for (int k = 0; k < RNN_; k += 4) {
        v2f a, b;
        a.x = hid[(size_t)M * RNN_ + k + koff];
        a.y = hid[(size_t)M * RNN_ + k + koff + 1];
        b.x = Wq[(size_t)N * RNN_ + k + koff];
        b.y = Wq[(size_t)N * RNN_ + k + koff + 1];
        c = __builtin_amdgcn_wmma_f32_16x16x4_f32(false, a, false, b,
                                                  (short)0, c, false, false);
    }
#pragma unroll
    for (int r = 0; r < 8; ++r) {
        const int row = mtile * 16 + r + half * 8;
        const int col = ntile * 16 + l;
        pq[row * ATT_ + col] = c[r];
    }
}

// ---------------------------------------------------------------------------
// Kernel 2: fused location conv (2->32, k=31) -> dense (32->128) via WMMA
//           -> tanh(pq + loc + processed_memory) . v  -> energies (B,T)
// Block = (b, 128-wide t tile), 256 threads = 8 waves (one M-tile per wave).
// ---------------------------------------------------------------------------
__global__ __launch_bounds__(256) void k_energy(
    const float* __restrict__ awc,   // (B,2,T)
    const float* __restrict__ cw,    // (32,2,31)
    const float* __restrict__ Wd,    // (128,32)
    const float* __restrict__ pq,    // (B,128)
    const float* __restrict__ pm,    // (B,T,128)
    const float* __restrict__ v,     // (128)
    float* __restrict__ energies)    // (B,T)
{
    __shared__ float cwS[NF_][2][KS_];      // 1984 f32
    __shared__ float WdS[ATT_][NF_ + 1];    // padded: no bank conflicts
    __shared__ float convS[128][NF_ + 1];   // conv output tile, padded

    const int b   = blockIdx.y;
    const int t0  = blockIdx.x * 128;
    const int tid = threadIdx.x;

    for (int i = tid; i < NF_ * 2 * KS_; i += 256)
        ((float*)cwS)[i] = cw[i];
    for (int i = tid; i < ATT_ * NF_; i += 256)
        WdS[i >> 5][i & 31] = Wd[i];
    __syncthreads();

    // ---- stage 1: location conv into LDS (each thread: 1 t, 16 filters) ----
    {
        const int tl = tid & 127;
        const int fb = (tid >> 7) * 16;
        float acc[16];
#pragma unroll
        for (int i = 0; i < 16; ++i) acc[i] = 0.f;
        for (int c = 0; c < 2; ++c) {
            const float* src = awc + (size_t)b * 2 * T_ + (size_t)c * T_;
            for (int k = 0; k < KS_; ++k) {
                const int gt = t0 + tl + k - PAD_;
                const float x = (gt >= 0 && gt < T_) ? src[gt] : 0.f;
#pragma unroll
                for (int i = 0; i < 16; ++i)
                    acc[i] = fmaf(x, cwS[fb + i][c][k], acc[i]);
            }
        }
#pragma unroll
        for (int i = 0; i < 16; ++i) convS[tl][fb + i] = acc[i];
    }
    __syncthreads();

    // ---- stage 2: WMMA dense + fused epilogue --------------------------------
    const int wave = tid >> 5;    // M tile (16 t values)
    const int lane = tid & 31;
    const int half = lane >> 4;
    const int l    = lane & 15;
    const int koff = half * 2;
    const int mrow = wave * 16 + l;   // A-matrix row for this lane

    float epart[8];
#pragma unroll
    for (int r = 0; r < 8; ++r) epart[r] = 0.f;

    for (int nt = 0; nt < 8; ++nt) {
        const int ncol = nt * 16 + l;
        v8f cacc = {};
#pragma unroll
        for (int kb = 0; kb < 8; ++kb) {
            const int k = kb * 4 + koff;
            v2f a, bb;
            a.x  = convS[mrow][k];
            a.y  = convS[mrow][k + 1];
            bb.x = WdS[ncol][k];
            bb.y = WdS[ncol][k + 1];
            cacc = __builtin_amdgcn_wmma_f32_16x16x4_f32(false, a, false, bb,
                                                         (short)0, cacc, false, false);
        }
        const float pqv = pq[b * ATT_ + ncol];
        const float vv  = v[ncol];
#pragma unroll
        for (int r = 0; r < 8; ++r) {
            const int t = t0 + wave * 16 + r + half * 8;
            const float p = __builtin_nontemporal_load(
                &pm[((size_t)b * T_ + t) * ATT_ + ncol]);
            epart[r] += tanhf(cacc[r] + pqv + p) * vv;
        }
    }

    // reduce over the 16 lanes of each half-wave (covers all 128 'a' columns)
#pragma unroll
    for (int r = 0; r < 8; ++r) {
        float s = epart[r];
        s += __shfl_xor(s, 1, 32);
        s += __shfl_xor(s, 2, 32);
        s += __shfl_xor(s, 4, 32);
        s += __shfl_xor(s, 8, 32);
        if (l == 0) {
            const int t = t0 + wave * 16 + r + half * 8;
            energies[(size_t)b * T_ + t] = s;
        }
    }
}

// ---------------------------------------------------------------------------
// Kernel 3: per-batch softmax over T, writes attention weights to d_out.
// ---------------------------------------------------------------------------
__global__ __launch_bounds__(256) void k_softmax(
    const float* __restrict__ energies,
    const unsigned char* __restrict__ mask,
    float* __restrict__ wout)            // (B,T)
{
    __shared__ float red[256];
    const int b = blockIdx.x, tid = threadIdx.x;
    const float* e = energies + (size_t)b * T_;
    const unsigned char* m = mask + (size_t)b * T_;

    float mx = -FLT_MAX;
    for (int t = tid; t < T_; t += 256)
        mx = fmaxf(mx, m[t] ? -FLT_MAX : e[t]);
    red[tid] = mx; __syncthreads();
    for (int s = 128; s; s >>= 1) {
        if (tid < s) red[tid] = fmaxf(red[tid], red[tid + s]);
        __syncthreads();
    }
    mx = red[0]; __syncthreads();

    float sum = 0.f;
    for (int t = tid; t < T_; t += 256)
        sum += m[t] ? 0.f : __expf(e[t] - mx);
    red[tid] = sum; __syncthreads();
    for (int s = 128; s; s >>= 1) {
        if (tid < s) red[tid] += red[tid + s];
        __syncthreads();
    }
    const float inv = 1.f / red[0];

    for (int t = tid; t < T_; t += 256)
        wout[(size_t)b * T_ + t] = m[t] ? 0.f : __expf(e[t] - mx) * inv;
}

__global__ void k_zero(float* __restrict__ p, int n) {
    int i = blockIdx.x * 256 + threadIdx.x;
    if (i < n) p[i] = 0.f;
}

// ---------------------------------------------------------------------------
// Kernel 4: context[b,:] = sum_t w[b,t] * memory[b,t,:]
// TDM (tensor_load_to_lds) double-buffers 16x512 f32 tiles of `memory` into
// LDS; wave 0 drives the DMA (TENSORcnt), all 8 waves consume after barrier.
// Partials combined with global_atomic_add_f32. Grid (T/128, B).
// ---------------------------------------------------------------------------
#define CTX_ROWS 16
#define CTX_TILES (128 / CTX_ROWS)

__global__ __launch_bounds__(256) void k_context(
    const float* __restrict__ memory,   // (B,T,EMB)
    const float* __restrict__ w,        // (B,T)
    float* __restrict__ ctx)            // (B,EMB), pre-zeroed
{
    __shared__ float buf[2][CTX_ROWS * EMB_];   // 2 x 32 KB double buffer
    __shared__ float wS[128];

    const int b   = blockIdx.y;
    const int t0  = blockIdx.x * 128;
    const int tid = threadIdx.x;

    if (tid < 128) wS[tid] = w[(size_t)b * T_ + t0 + tid];

    const unsigned lds0 = (unsigned)(size_t)&buf[0][0];
    const unsigned lds1 = (unsigned)(size_t)&buf[1][0];

    // D# group 1 (constant): 16x512 f32 tile == tensor; row stride 512 elems.
    i32x8 g1;
    g1[0] = 0x00020000;            // data_size=4B; no multicast/pad/iterate
    g1[1] = (int)(512u << 16);     // tensor_dim0 = 512 (bits 63:48)
    g1[2] = (int)(16u << 16);      // tensor_dim1 = 16  (bits 95:80)
    g1[3] = (int)(512u << 16);     // tile_dim0  = 512 (bits 127:112)
    g1[4] = 16;                    // tile_dim1  = 16  (bits 143:128)
    g1[5] = 512;                   // tensor_dim0_stride[31:0] = 512
    g1[6] = 0;
    g1[7] = 0;
    const i32x4 gz4 = {0, 0, 0, 0};            // groups 2/3 unused (2-D tile)
    const i32x8 gz8 = {0, 0, 0, 0, 0, 0, 0, 0};

    const bool issuer = (tid < 32);   // wave 0 programs the TDM

    auto issue = [&](int tile, unsigned lds) {
        unsigned long long ga = (unsigned long long)(size_t)
            (memory + ((size_t)b * T_ + t0 + tile * CTX_ROWS) * EMB_);
        u32x4 g0;
        g0[0] = 1u;                          // count=1 valid user descriptor
        g0[1] = lds;                         // LDS byte address
        g0[2] = (unsigned)ga;                // global_addr[31:0]
        g0[3] = (unsigned)((ga >> 32) & 0x01FFFFFFu) | 0x80000000u; // [56:32]|type=2
        __builtin_amdgcn_tensor_load_to_lds(g0, g1, gz4, gz4, gz8, 0);
    };

    if (issuer) issue(0, lds0);

    v2f acc; acc.x = 0.f; acc.y = 0.f;
    for (int tile = 0; tile < CTX_TILES; ++tile) {
        if (issuer) {
            if (tile + 1 < CTX_TILES) {
                issue(tile + 1, ((tile + 1) & 1) ? lds1 : lds0);
                __builtin_amdgcn_s_wait_tensorcnt(1);  // current tile resident
            } else {
                __builtin_amdgcn_s_wait_tensorcnt(0);  // last tile resident
            }
        }
        __syncthreads();                      // tile visible to all waves
        const float* src = &buf[tile & 1][0];
#pragma unroll
        for (int r = 0; r < CTX_ROWS; ++r) {
            const float wt = wS[tile * CTX_ROWS + r];
            v2f mv = ((const v2f*)(src + r * EMB_))[tid];
            acc.x = fmaf(wt, mv.x, acc.x);
            acc.y = fmaf(wt, mv.y, acc.y);
        }
        __syncthreads();                      // reads done before buffer reuse
    }
    atomicAdd(&ctx[b * EMB_ + 2 * tid],     acc.x);
    atomicAdd(&ctx[b * EMB_ + 2 * tid + 1], acc.y);
}

// ---------------------------------------------------------------------------
extern "C" void kernel_launch(void* const* d_in, const int* in_sizes, int n_in,
                              void* d_out, int out_size, void* d_ws, size_t ws_size,
                              hipStream_t stream) {
    const float* hid = (const float*)d_in[0];          // (B,RNN)
    const float* mem = (const float*)d_in[1];          // (B,T,EMB)
    const float* pm  = (const float*)d_in[2];          // (B,T,ATT)
    const float* awc = (const float*)d_in[3];          // (B,2,T)
    const unsigned char* mask = (const unsigned char*)d_in[4]; // (B,T) bool
    const float* Wq  = (const float*)d_in[5];          // (ATT,RNN)
    const float* cw  = (const float*)d_in[6];          // (NF,2,KS)
    const float* Wd  = (const float*)d_in[7];          // (ATT,NF)
    const float* v   = (const float*)d_in[8];          // (1,ATT)

    float* out_ctx = (float*)d_out;                    // (B,EMB)
    float* out_w   = out_ctx + (size_t)B_ * EMB_;      // (B,T)

    float* ws_energy = (float*)d_ws;                   // B*T f32
    float* ws_pq     = ws_energy + (size_t)B_ * T_;    // B*ATT f32

    // 1) query projection via WMMA f32 16x16x4
    k_pq<<<dim3(32), dim3(32), 0, stream>>>(hid, Wq, ws_pq);

    // 2) fused conv + WMMA dense + tanh + v-dot -> energies
    k_energy<<<dim3(T_ / 128, B_), dim3(256), 0, stream>>>(
        awc, cw, Wd, ws_pq, pm, v, ws_energy);

    // 3) softmax -> attention weights (second output)
    k_softmax<<<dim3(B_), dim3(256), 0, stream>>>(ws_energy, mask, out_w);

    // 4) context = weights @ memory (first output), TDM-staged
    k_zero<<<dim3((B_ * EMB_ + 255) / 256), dim3(256), 0, stream>>>(out_ctx, B_ * EMB_);
    k_context<<<dim3(T_ / 128, B_), dim3(256), 0, stream>>>(mem, out_w, out_ctx);
}